// NeuralODEEncoderRNN_61040075210859
// MI455X (gfx1250) — compile-verified
//
#include <hip/hip_runtime.h>
#include <math.h>

typedef float v2f __attribute__((ext_vector_type(2)));
typedef float v8f __attribute__((ext_vector_type(8)));

#define B_      512
#define H_      256
#define NSTATE  (B_ * H_)
#define MAXSTEP 64

// ---------------------------------------------------------------------------
// fp32 WMMA helper: D = A(16x4) * B(4x16) + C(16x16), full fp32 accumulate.
// ---------------------------------------------------------------------------
__device__ __forceinline__ v8f wmma_f32(v2f a, v2f b, v8f c) {
    return __builtin_amdgcn_wmma_f32_16x16x4_f32(
        /*neg_a=*/false, a, /*neg_b=*/false, b,
        /*c_mod=*/(short)0, c, /*reuse_a=*/false, /*reuse_b=*/false);
}

// ---------------------------------------------------------------------------
// Generic bias-GEMM: out[m, n] = sum_k A[m, k] * W[n, k] + bias[n]
// One wave (32 threads) computes one 16x16 output tile via V_WMMA_F32_16X16X4.
// ---------------------------------------------------------------------------
__global__ void gemm_bias_kernel(const float* __restrict__ A, int lda,
                                 const float* __restrict__ W, int ldw,
                                 const float* __restrict__ bias,
                                 float* __restrict__ out, int ldo, int K) {
    const int lane = threadIdx.x;
    const int m0   = blockIdx.x * 16;
    const int n0   = blockIdx.y * 16;
    const int rc   = lane & 15;            // row for A frag / col for B,C frag
    const int kb   = (lane >> 4) << 1;     // K sub-offset: 0 or 2

    const float* arow = A + (size_t)(m0 + rc) * lda;
    const float* wrow = W + (size_t)(n0 + rc) * ldw;

    v8f acc = {};
    for (int k = 0; k < K; k += 4) {
        v2f a, b;
        a.x = arow[k + kb];  a.y = arow[k + kb + 1];
        b.x = wrow[k + kb];  b.y = wrow[k + kb + 1];
        acc = wmma_f32(a, b, acc);
    }

    const float bv    = bias[n0 + rc];
    const int   rbase = m0 + ((lane >> 4) << 3);   // rows r / 8+r per C layout
#pragma unroll
    for (int r = 0; r < 8; ++r)
        out[(size_t)(rbase + r) * ldo + (n0 + rc)] = acc[r] + bv;
}

// ---------------------------------------------------------------------------
// GRU-gate ODE function evaluation (one dopri5 stage), templated on the
// number of accumulated k-buffers so the stage combine is fully unrolled and
// the K-loop body is straight-line (clause-friendly, software-pipelineable).
//
//   A      = base + dt_c * sum_j c_j * k_j         (built on the fly)
//   G      = A @ Wih[:, :256]^T                     (3 gate column groups)
//   t_eval = t + cstage*dt_c  folded into bias via Wih[:,256]
//   k_out  = (1 - sigmoid(Gz+bz)) * tanh(Gn + sigmoid(Gr+br)*bhh_n)
//
// One wave per 32(batch) x 16(hidden) tile: 2 M-subtiles x 3 gates =
// 6 WMMA accumulator chains; each B fragment feeds 2 WMMAs.
// ---------------------------------------------------------------------------
struct KAcc {
    const float* p0; const float* p1; const float* p2;
    const float* p3; const float* p4;
    float c0, c1, c2, c3, c4;
};

template <int NACC>
__global__ void feval_kernel(const float* __restrict__ base,  // 512x256
                             KAcc acc,
                             const float* __restrict__ Wih,   // 768x257
                             const float* __restrict__ bih,   // 768
                             const float* __restrict__ bhh,   // 768
                             const float* __restrict__ ctrl,
                             float cstage,
                             float* __restrict__ Kout) {      // 512x256
    const int lane = threadIdx.x;
    const int m0   = blockIdx.x * 32;     // 32 batch rows per wave
    const int h0   = blockIdx.y * 16;
    const int rc   = lane & 15;
    const int kb   = (lane >> 4) << 1;

    const float t     = ctrl[0];
    const float dtc   = ctrl[2];
    const float teval = t + cstage * dtc;

    const int n_r = h0 + rc;
    const int n_z = 256 + n_r;
    const int n_n = 512 + n_r;
    const float* wr = Wih + (size_t)n_r * 257;
    const float* wz = Wih + (size_t)n_z * 257;
    const float* wn = Wih + (size_t)n_n * 257;

    const size_t row0 = (size_t)(m0 + rc) * H_;
    const size_t row1 = (size_t)(m0 + 16 + rc) * H_;
    const float* arow0 = base + row0;
    const float* arow1 = base + row1;

    const float* kptr[5] = {acc.p0, acc.p1, acc.p2, acc.p3, acc.p4};
    const float  kcf[5]  = {acc.c0, acc.c1, acc.c2, acc.c3, acc.c4};
    const float* kr0[NACC > 0 ? NACC : 1];
    const float* kr1[NACC > 0 ? NACC : 1];
    float        kc[NACC > 0 ? NACC : 1];
#pragma unroll
    for (int j = 0; j < NACC; ++j) {
        kr0[j] = kptr[j] + row0;
        kr1[j] = kptr[j] + row1;
        kc[j]  = kcf[j] * dtc;
    }

    v8f cr0 = {}, cr1 = {}, cz0 = {}, cz1 = {}, cn0 = {}, cn1 = {};
    for (int k = 0; k < H_; k += 4) {
        const int kk = k + kb;
        v2f a0, a1;
        a0.x = arow0[kk];  a0.y = arow0[kk + 1];
        a1.x = arow1[kk];  a1.y = arow1[kk + 1];
#pragma unroll
        for (int j = 0; j < NACC; ++j) {
            a0.x += kc[j] * kr0[j][kk];
            a0.y += kc[j] * kr0[j][kk + 1];
            a1.x += kc[j] * kr1[j][kk];
            a1.y += kc[j] * kr1[j][kk + 1];
        }
        v2f br_, bz_, bn_;
        br_.x = wr[kk];  br_.y = wr[kk + 1];
        bz_.x = wz[kk];  bz_.y = wz[kk + 1];
        bn_.x = wn[kk];  bn_.y = wn[kk + 1];
        cr0 = wmma_f32(a0, br_, cr0);
        cr1 = wmma_f32(a1, br_, cr1);
        cz0 = wmma_f32(a0, bz_, cz0);
        cz1 = wmma_f32(a1, bz_, cz1);
        cn0 = wmma_f32(a0, bn_, cn0);
        cn1 = wmma_f32(a1, bn_, cn1);
    }

    // bias with time column folded in: bih + t_eval * Wih[:,256] (+ bhh)
    const float biasr = bih[n_r] + teval * wr[256] + bhh[n_r];
    const float biasz = bih[n_z] + teval * wz[256] + bhh[n_z];
    const float biasn = bih[n_n] + teval * wn[256];
    const float bnh   = bhh[n_n];

    const int rbase0 = m0 + ((lane >> 4) << 3);
    const int rbase1 = rbase0 + 16;
#pragma unroll
    for (int r = 0; r < 8; ++r) {
        {
            const float gr = cr0[r] + biasr;
            const float gz = cz0[r] + biasz;
            const float gn = cn0[r] + biasn;
            const float rg = 1.f / (1.f + expf(-gr));
            const float zg = 1.f / (1.f + expf(-gz));
            const float ng = tanhf(gn + rg * bnh);
            Kout[(size_t)(rbase0 + r) * H_ + n_r] = (1.f - zg) * ng;
        }
        {
            const float gr = cr1[r] + biasr;
            const float gz = cz1[r] + biasz;
            const float gn = cn1[r] + biasn;
            const float rg = 1.f / (1.f + expf(-gr));
            const float zg = 1.f / (1.f + expf(-gz));
            const float ng = tanhf(gn + rg * bnh);
            Kout[(size_t)(rbase1 + r) * H_ + n_r] = (1.f - zg) * ng;
        }
    }
}

// ---------------------------------------------------------------------------
// Control / elementwise kernels.
// ctrl layout: [0]=t [1]=dt [2]=dt_c [3]=done [4]=err_sum [5]=commit [6]=t1
// ---------------------------------------------------------------------------
__global__ void init_ctrl_kernel(const float* __restrict__ tp, int ntp,
                                 float* __restrict__ ctrl) {
    const float t0 = tp[0], t1 = tp[ntp - 1];
    ctrl[0] = t0;
    ctrl[1] = (t1 - t0) * 0.01f + 1e-8f;
    ctrl[6] = t1;
}

__global__ void ctrl_begin_kernel(float* __restrict__ ctrl) {
    const float t = ctrl[0], dt = ctrl[1], t1 = ctrl[6];
    const float rem = t1 - t;
    ctrl[3] = (rem <= 1e-6f) ? 1.f : 0.f;   // done
    ctrl[2] = fminf(dt, rem);               // dt_c
    ctrl[4] = 0.f;                          // zero err accumulator
}

__global__ void combine5_kernel(const float* __restrict__ Y,
                                const float* __restrict__ k1,
                                const float* __restrict__ k3,
                                const float* __restrict__ k4,
                                const float* __restrict__ k5,
                                const float* __restrict__ k6,
                                const float* __restrict__ ctrl,
                                float* __restrict__ Ynew) {
    const int   i   = blockIdx.x * blockDim.x + threadIdx.x;
    const float dtc = ctrl[2];
    Ynew[i] = Y[i] + dtc * ((35.f / 384.f) * k1[i] + (500.f / 1113.f) * k3[i] +
                            (125.f / 192.f) * k4[i] +
                            (-2187.f / 6784.f) * k5[i] + (11.f / 84.f) * k6[i]);
}

__global__ void err_reduce_kernel(const float* __restrict__ k1,
                                  const float* __restrict__ k3,
                                  const float* __restrict__ k4,
                                  const float* __restrict__ k5,
                                  const float* __restrict__ k6,
                                  const float* __restrict__ k7,
                                  const float* __restrict__ Y,
                                  const float* __restrict__ Ynew,
                                  float* __restrict__ ctrl) {
    __shared__ float s[256];
    const int   i   = blockIdx.x * blockDim.x + threadIdx.x;
    const float dtc = ctrl[2];
    const float e = dtc * ((71.f / 57600.f) * k1[i] + (-71.f / 16695.f) * k3[i] +
                           (71.f / 1920.f) * k4[i] +
                           (-17253.f / 339200.f) * k5[i] +
                           (22.f / 525.f) * k6[i] + (-1.f / 40.f) * k7[i]);
    const float sc = 1e-6f + 1e-5f * fmaxf(fabsf(Y[i]), fabsf(Ynew[i]));
    float q = e / sc;
    q *= q;
    s[threadIdx.x] = q;
    __syncthreads();
    for (int off = 128; off > 0; off >>= 1) {
        if (threadIdx.x < off) s[threadIdx.x] += s[threadIdx.x + off];
        __syncthreads();
    }
    if (threadIdx.x == 0) atomicAdd(ctrl + 4, s[0]);
}

__global__ void ctrl_end_kernel(float* __restrict__ ctrl) {
    const float t = ctrl[0], dt = ctrl[1], dtc = ctrl[2];
    const float done = ctrl[3], es = ctrl[4];
    const float err_norm = sqrtf(es / (float)NSTATE);
    const bool  accept   = err_norm <= 1.f;
    float factor = 0.9f * powf(err_norm + 1e-10f, -0.2f);
    factor       = fminf(fmaxf(factor, 0.2f), 10.f);
    const float t_n  = accept ? (t + dtc) : t;
    const float dt_n = dtc * factor;
    ctrl[0] = (done > 0.5f) ? t : t_n;
    ctrl[1] = (done > 0.5f) ? dt : dt_n;
    ctrl[5] = (done < 0.5f && accept) ? 1.f : 0.f;   // commit flag
}

__global__ void commit_kernel(float* __restrict__ Y,
                              const float* __restrict__ Ynew,
                              const float* __restrict__ ctrl) {
    const int i = blockIdx.x * blockDim.x + threadIdx.x;
    if (ctrl[5] > 0.5f) Y[i] = Ynew[i];
}

// ---------------------------------------------------------------------------
// Launch sequence (graph-capturable, deterministic fixed 64-step schedule).
// ---------------------------------------------------------------------------
extern "C" void kernel_launch(void* const* d_in, const int* in_sizes, int n_in,
                              void* d_out, int out_size, void* d_ws,
                              size_t ws_size, hipStream_t stream) {
    (void)n_in; (void)out_size; (void)ws_size;
    const float* x   = (const float*)d_in[0];   // 512x128
    const float* tp  = (const float*)d_in[1];   // 8
    const float* Wp  = (const float*)d_in[2];   // 256x128
    const float* bp  = (const float*)d_in[3];   // 256
    const float* Wih = (const float*)d_in[4];   // 768x257
    const float* bih = (const float*)d_in[5];   // 768
    /* d_in[6] = Whh: unused (zero initial hidden state) */
    const float* bhh = (const float*)d_in[7];   // 768
    const float* Wl  = (const float*)d_in[8];   // 64x256
    const float* bl  = (const float*)d_in[9];   // 64
    float*       out = (float*)d_out;           // 512x64

    float* Y    = (float*)d_ws;
    float* Ynew = Y + NSTATE;
    float* k1   = Ynew + NSTATE;
    float* k2   = k1 + NSTATE;
    float* k3   = k2 + NSTATE;
    float* k4   = k3 + NSTATE;
    float* k5   = k4 + NSTATE;
    float* k6   = k5 + NSTATE;
    float* k7   = k6 + NSTATE;
    float* ctrl = k7 + NSTATE;

    const dim3 fgrid(B_ / 32, H_ / 16);   // 16 x 16 tiles, 1 wave each

    init_ctrl_kernel<<<1, 1, 0, stream>>>(tp, in_sizes[1], ctrl);
    // h0 = x @ Wp^T + bp
    gemm_bias_kernel<<<dim3(B_ / 16, H_ / 16), 32, 0, stream>>>(
        x, 128, Wp, 128, bp, Y, H_, 128);

    for (int step = 0; step < MAXSTEP; ++step) {
        ctrl_begin_kernel<<<1, 1, 0, stream>>>(ctrl);

        KAcc a;
        a = {k1, k1, k1, k1, k1, 0.f, 0.f, 0.f, 0.f, 0.f};
        feval_kernel<0><<<fgrid, 32, 0, stream>>>(Y, a, Wih, bih, bhh, ctrl,
                                                  0.f, k1);
        a = {k1, k1, k1, k1, k1, 0.2f, 0.f, 0.f, 0.f, 0.f};
        feval_kernel<1><<<fgrid, 32, 0, stream>>>(Y, a, Wih, bih, bhh, ctrl,
                                                  0.2f, k2);
        a = {k1, k2, k1, k1, k1, 3.f / 40.f, 9.f / 40.f, 0.f, 0.f, 0.f};
        feval_kernel<2><<<fgrid, 32, 0, stream>>>(Y, a, Wih, bih, bhh, ctrl,
                                                  0.3f, k3);
        a = {k1, k2, k3, k1, k1, 44.f / 45.f, -56.f / 15.f, 32.f / 9.f, 0.f,
             0.f};
        feval_kernel<3><<<fgrid, 32, 0, stream>>>(Y, a, Wih, bih, bhh, ctrl,
                                                  0.8f, k4);
        a = {k1, k2, k3, k4, k1, 19372.f / 6561.f, -25360.f / 2187.f,
             64448.f / 6561.f, -212.f / 729.f, 0.f};
        feval_kernel<4><<<fgrid, 32, 0, stream>>>(Y, a, Wih, bih, bhh, ctrl,
                                                  8.f / 9.f, k5);
        a = {k1, k2, k3, k4, k5, 9017.f / 3168.f, -355.f / 33.f,
             46732.f / 5247.f, 49.f / 176.f, -5103.f / 18656.f};
        feval_kernel<5><<<fgrid, 32, 0, stream>>>(Y, a, Wih, bih, bhh, ctrl,
                                                  1.f, k6);

        combine5_kernel<<<B_, H_, 0, stream>>>(Y, k1, k3, k4, k5, k6, ctrl,
                                               Ynew);

        a = {k1, k1, k1, k1, k1, 0.f, 0.f, 0.f, 0.f, 0.f};
        feval_kernel<0><<<fgrid, 32, 0, stream>>>(Ynew, a, Wih, bih, bhh, ctrl,
                                                  1.f, k7);

        err_reduce_kernel<<<B_, H_, 0, stream>>>(k1, k3, k4, k5, k6, k7, Y,
                                                 Ynew, ctrl);
        ctrl_end_kernel<<<1, 1, 0, stream>>>(ctrl);
        commit_kernel<<<B_, H_, 0, stream>>>(Y, Ynew, ctrl);
    }

    // out = hT @ Wl^T + bl
    gemm_bias_kernel<<<dim3(B_ / 16, 64 / 16), 32, 0, stream>>>(
        Y, H_, Wl, H_, bl, out, 64, H_);
}